// SimpleNet_62457414419059
// MI455X (gfx1250) — compile-verified
//
#include <hip/hip_runtime.h>
#include <math.h>

// ---------------------------------------------------------------------------
// Quaternion tangent-norm reduction for MI455X (gfx1250).
// Bandwidth-bound map-reduce: stream a[] and idd[] (float4 each) through the
// CDNA5 async global->LDS pipe (ASYNCcnt), per-wave double buffered, then
// deterministic wave32 tree reduction -> per-block partials -> final sqrt.
// ---------------------------------------------------------------------------

#define TILE   256   // quats per tile  (== BLOCK: each lane stages its own slot)
#define BLOCK  256   // 8 wave32 waves
#define MAX_BLOCKS 2048

// Flat shared-aperture address: addr[63:32] = SHARED_BASE, addr[31:0] = LDS
// byte offset (CDNA5 ISA §aperture).  Low 32 bits are the DS address.
__device__ __forceinline__ unsigned lds_u32(const void* p) {
  return (unsigned)(uintptr_t)p;
}

// GLOBAL_LOAD_ASYNC_TO_LDS_B128, GVS mode: mem = SADDR(64) + VADDR(i32) + 0.
// vdst operand carries the LDS byte address.  Tracked by ASYNCcnt.
__device__ __forceinline__ void async_copy_b128(unsigned lds_byte,
                                                unsigned long long gbase,
                                                int byte_off) {
  asm volatile("global_load_async_to_lds_b128 %0, %1, %2"
               :
               : "v"(lds_byte), "v"(byte_off), "s"(gbase)
               : "memory");
}

__device__ __forceinline__ void wait_async_le2() {
  asm volatile("s_wait_asynccnt 0x2" ::: "memory");
}
__device__ __forceinline__ void wait_async_0() {
  asm volatile("s_wait_asynccnt 0x0" ::: "memory");
}

// Per-element contribution to ||tangent||^2.
// rel = qmul(qinv(a), b):  w = w1*w2 + v1.v2 ;  v = w1*v2 - w2*v1 - v1 x v2
// contribution = (nv < eps) ? nv^2 * (2/w)^2 : (2*atan2(nv, w))^2
__device__ __forceinline__ float qcontrib(float4 qa, float4 qb) {
  const float w1 = qa.x, x1 = qa.y, y1 = qa.z, z1 = qa.w;
  const float w2 = qb.x, x2 = qb.y, y2 = qb.z, z2 = qb.w;
  const float w  = w1 * w2 + x1 * x2 + y1 * y2 + z1 * z2;
  const float vx = w1 * x2 - w2 * x1 - (y1 * z2 - z1 * y2);
  const float vy = w1 * y2 - w2 * y1 - (z1 * x2 - x1 * z2);
  const float vz = w1 * z2 - w2 * z1 - (x1 * y2 - y1 * x2);
  const float nv2 = vx * vx + vy * vy + vz * vz;
  const float nv  = sqrtf(nv2);
  if (nv < 1e-6f) {
    const float s = 2.0f / w;
    return nv2 * s * s;
  }
  const float theta = 2.0f * atan2f(nv, w);
  return theta * theta;
}

__global__ __launch_bounds__(BLOCK) void quat_tan_partials(
    const float4* __restrict__ a, const float4* __restrict__ b,
    float* __restrict__ partials, int n) {
  __shared__ float4 sA[2][TILE];
  __shared__ float4 sB[2][TILE];
  __shared__ float  wred[BLOCK / 32];

  const int tid = threadIdx.x;
  const unsigned ldsA[2] = {lds_u32(&sA[0][tid]), lds_u32(&sA[1][tid])};
  const unsigned ldsB[2] = {lds_u32(&sB[0][tid]), lds_u32(&sB[1][tid])};
  const unsigned long long ga = (unsigned long long)(uintptr_t)a;
  const unsigned long long gb = (unsigned long long)(uintptr_t)b;

  const int numTiles = n / TILE;
  const int stride   = (int)gridDim.x;
  float acc = 0.0f;

  // ---- async double-buffered stream over full tiles --------------------
  int tile = (int)blockIdx.x;
  int buf  = 0;
  if (tile < numTiles) {
    const int off = (tile * TILE + tid) * 16;
    async_copy_b128(ldsA[0], ga, off);
    async_copy_b128(ldsB[0], gb, off);
  }
  while (tile < numTiles) {
    const int next = tile + stride;
    if (next < numTiles) {
      const int off = (next * TILE + tid) * 16;
      async_copy_b128(ldsA[buf ^ 1], ga, off);
      async_copy_b128(ldsB[buf ^ 1], gb, off);
      wait_async_le2();   // this tile's 2 copies done; next tile's 2 in flight
    } else {
      wait_async_0();
    }
    // TILE == BLOCK: each lane consumes exactly the slot it staged, so only
    // the per-wave ASYNCcnt wait is needed (no block barrier in the loop).
    const float4 qa = sA[buf][tid];
    const float4 qb = sB[buf][tid];
    acc += qcontrib(qa, qb);
    buf ^= 1;
    tile = next;
  }

  // ---- tail (n % TILE != 0; empty for N = 4,000,000) -------------------
  const int tailStart = numTiles * TILE;
  for (int i = tailStart + (int)blockIdx.x * BLOCK + tid; i < n;
       i += stride * BLOCK) {
    acc += qcontrib(a[i], b[i]);
  }

  // ---- deterministic block reduction (wave32) --------------------------
#pragma unroll
  for (int m = 16; m >= 1; m >>= 1) acc += __shfl_xor(acc, m, 32);
  const int lane = tid & 31;
  const int wave = tid >> 5;
  if (lane == 0) wred[wave] = acc;
  __syncthreads();
  if (wave == 0) {
    float v = (lane < BLOCK / 32) ? wred[lane] : 0.0f;
#pragma unroll
    for (int m = 4; m >= 1; m >>= 1) v += __shfl_xor(v, m, 32);
    if (lane == 0) partials[blockIdx.x] = v;
  }
}

__global__ __launch_bounds__(BLOCK) void quat_tan_finalize(
    const float* __restrict__ partials, int np, float* __restrict__ out) {
  __shared__ float wred[BLOCK / 32];
  const int tid = threadIdx.x;
  float acc = 0.0f;
  for (int i = tid; i < np; i += BLOCK) acc += partials[i];
#pragma unroll
  for (int m = 16; m >= 1; m >>= 1) acc += __shfl_xor(acc, m, 32);
  const int lane = tid & 31;
  const int wave = tid >> 5;
  if (lane == 0) wred[wave] = acc;
  __syncthreads();
  if (wave == 0) {
    float v = (lane < BLOCK / 32) ? wred[lane] : 0.0f;
#pragma unroll
    for (int m = 4; m >= 1; m >>= 1) v += __shfl_xor(v, m, 32);
    if (lane == 0) out[0] = sqrtf(v);
  }
}

extern "C" void kernel_launch(void* const* d_in, const int* in_sizes, int n_in,
                              void* d_out, int out_size, void* d_ws,
                              size_t ws_size, hipStream_t stream) {
  const float4* a = (const float4*)d_in[0];
  const float4* b = (const float4*)d_in[1];
  const int n = in_sizes[0] / 4;  // quaternion count

  int blocks = MAX_BLOCKS;
  const int wsCap = (int)(ws_size / sizeof(float));
  if (blocks > wsCap) blocks = wsCap;
  if (blocks < 1) blocks = 1;

  float* partials = (float*)d_ws;  // fully overwritten each call: deterministic
  quat_tan_partials<<<blocks, BLOCK, 0, stream>>>(a, b, partials, n);
  quat_tan_finalize<<<1, BLOCK, 0, stream>>>(partials, blocks, (float*)d_out);
}